// SpatialOut_84387517432009
// MI455X (gfx1250) — compile-verified
//
#include <hip/hip_runtime.h>
#include <hip/hip_bf16.h>

typedef float v2f __attribute__((ext_vector_type(2)));
typedef float v8f __attribute__((ext_vector_type(8)));

// ---- LDS layout (floats) ----
// W1^T  : 64 rows x stride 132  (pad: stride%64==4 -> conflict-free column reads)
// Wsph^T: 32 rows x stride 68
// per-wave scalar tile: 16 x stride 132
// per-wave sph    tile: 16 x stride 196
#define OFF_W1T   0
#define W1T_STR   132
#define OFF_WST   (64 * W1T_STR)              // 8448
#define WST_STR   68
#define OFF_SA    (OFF_WST + 32 * WST_STR)    // 10624
#define SA_STR    132
#define SA_SZ     (16 * SA_STR)               // 2112
#define OFF_SS    (OFF_SA + 8 * SA_SZ)        // 27520
#define SS_STR    196
#define SS_SZ     (16 * SS_STR)               // 3136
#define SMEM_FLOATS (OFF_SS + 8 * SS_SZ)      // 52608 -> 210432 bytes (< 320KB WGP LDS)

#define LOG2E 1.44269504088896340736f

// fast sigmoid: v_exp_f32 + v_add + v_rcp_f32 (no IEEE divide sequences)
__device__ __forceinline__ float fast_sigmoid(float x) {
  return __builtin_amdgcn_rcpf(1.f + __builtin_amdgcn_exp2f(-LOG2E * x));
}

__device__ __forceinline__ void butterfly16(float v[8]) {
#pragma unroll
  for (int off = 1; off <= 8; off <<= 1) {
#pragma unroll
    for (int i = 0; i < 8; ++i) v[i] += __shfl_xor(v[i], off, 32);
  }
}

__device__ __forceinline__ float pick8(const float v[8], int i) {
  float x = v[0];
  x = (i == 1) ? v[1] : x;
  x = (i == 2) ? v[2] : x;
  x = (i == 3) ? v[3] : x;
  x = (i == 4) ? v[4] : x;
  x = (i == 5) ? v[5] : x;
  x = (i == 6) ? v[6] : x;
  x = (i == 7) ? v[7] : x;
  return x;
}

__global__ void __launch_bounds__(256, 1)
spatial_main(const float* __restrict__ xs, const float* __restrict__ xsph,
             const float* __restrict__ W1, const float* __restrict__ b1,
             const float* __restrict__ W2, const float* __restrict__ b2,
             const float* __restrict__ Wsph1, const float* __restrict__ Wsph2,
             float* __restrict__ atom_out, int N, int tiles) {
  extern __shared__ float sm[];
  const int tid  = threadIdx.x;
  const int lane = tid & 31;
  const int wid  = tid >> 5;
  const int r    = lane & 15;
  const int half = lane >> 4;
  const int kb   = half << 1;  // K sub-offset for this half-wave (A/B frag layout)

  // ---- stage weights to LDS, transposed + padded (once per block) ----
  for (int idx = tid; idx < 128 * 64; idx += 256) {
    int k = idx >> 6, n = idx & 63;
    sm[OFF_W1T + n * W1T_STR + k] = W1[idx];
  }
  for (int idx = tid; idx < 64 * 32; idx += 256) {
    int i = idx >> 5, j = idx & 31;
    sm[OFF_WST + j * WST_STR + i] = Wsph1[idx];
  }
  __syncthreads();

  // per-lane epilogue constants: this lane owns output column n = nt*16 + r
  float b1v[4], w2v[4], wsp2v[2];
#pragma unroll
  for (int nt = 0; nt < 4; ++nt) { b1v[nt] = b1[nt * 16 + r]; w2v[nt] = W2[nt * 16 + r]; }
#pragma unroll
  for (int nt = 0; nt < 2; ++nt) wsp2v[nt] = Wsph2[nt * 16 + r];
  const float b2s = b2[0];

  float* sA = &sm[OFF_SA + wid * SA_SZ];
  float* sS = &sm[OFF_SS + wid * SS_SZ];
  const float* w1t = &sm[OFF_W1T];
  const float* wst = &sm[OFF_WST];

  const v8f vz = {0.f, 0.f, 0.f, 0.f, 0.f, 0.f, 0.f, 0.f};

  const int gw0 = blockIdx.x * 8 + wid;
  const int gstep = gridDim.x * 8;

  for (int t = gw0; t < tiles; t += gstep) {
    const int a0 = t * 16;
    __builtin_amdgcn_wave_barrier();

    // ---- stage x_scalar tile 16x128 (coalesced b128) ----
#pragma unroll 4
    for (int i = 0; i < 16; ++i) {
      int row = a0 + i; if (row >= N) row = N - 1;
      float4 v = *(const float4*)(xs + (size_t)row * 128 + lane * 4);
      *(float4*)(sA + i * SA_STR + lane * 4) = v;
    }
    // ---- stage 64x1e slice 16x192 (cols 128..319 of x_spherical) ----
#pragma unroll 4
    for (int j = 0; j < 24; ++j) {
      int lin = j * 128 + lane * 4;
      int row = lin / 192, col = lin % 192;
      int grow = a0 + row; if (grow >= N) grow = N - 1;
      float4 v = *(const float4*)(xsph + (size_t)grow * 480 + 128 + col);
      *(float4*)(sS + row * SS_STR + col) = v;
    }
    asm volatile("s_wait_dscnt 0" ::: "memory");
    __builtin_amdgcn_wave_barrier();

    // ---- prefetch next grid-stride tile into cache (global_prefetch_b8) ----
    const int tn = t + gstep;
    if (tn < tiles) {
      const int an = tn * 16;
      // x_scalar: 16 rows * 512B = 64 cachelines; 2 per lane
#pragma unroll
      for (int q = 0; q < 2; ++q) {
        int lineid = lane + 32 * q;
        int prow = an + (lineid >> 2);
        if (prow < N)
          __builtin_prefetch(xs + (size_t)prow * 128 + (lineid & 3) * 32, 0, 1);
      }
      // sph slice: 16 rows * 768B = 96 cachelines; 3 per lane
#pragma unroll
      for (int q = 0; q < 3; ++q) {
        int lineid = lane + 32 * q;
        int prow = an + lineid / 6;
        if (prow < N)
          __builtin_prefetch(xsph + (size_t)prow * 480 + 128 + (lineid % 6) * 32, 0, 1);
      }
    }

    // ---- scalar GEMM: [16x128] @ [128x64] via v_wmma_f32_16x16x4_f32 ----
    v8f acc[4];
#pragma unroll
    for (int nt = 0; nt < 4; ++nt) acc[nt] = vz;
#pragma unroll 4
    for (int s = 0; s < 32; ++s) {
      const int k0 = 4 * s + kb;
      v2f a = *(const v2f*)(sA + r * SA_STR + k0);
#pragma unroll
      for (int nt = 0; nt < 4; ++nt) {
        v2f bf = *(const v2f*)(w1t + (nt * 16 + r) * W1T_STR + k0);
        acc[nt] = __builtin_amdgcn_wmma_f32_16x16x4_f32(
            false, a, false, bf, (short)0, acc[nt], false, false);
      }
    }

    // ---- scalar epilogue: +b1, SiLU, dot W2, reduce over 64 cols ----
    float p[8];
#pragma unroll
    for (int v = 0; v < 8; ++v) p[v] = 0.f;
#pragma unroll
    for (int nt = 0; nt < 4; ++nt)
#pragma unroll
      for (int v = 0; v < 8; ++v) {
        float y = acc[nt][v] + b1v[nt];
        p[v] += y * fast_sigmoid(y) * w2v[nt];
      }
    butterfly16(p);
    float so[8];
#pragma unroll
    for (int v = 0; v < 8; ++v) so[v] = p[v] + b2s;

    // ---- spherical GEMM: per component c, [16x64] @ [64x32] ----
    v8f hacc[3][2];
#pragma unroll
    for (int c = 0; c < 3; ++c) { hacc[c][0] = vz; hacc[c][1] = vz; }
#pragma unroll 2
    for (int s = 0; s < 16; ++s) {
      const int k0 = 4 * s + kb;
      v2f bb0 = *(const v2f*)(wst + r * WST_STR + k0);
      v2f bb1 = *(const v2f*)(wst + (16 + r) * WST_STR + k0);
#pragma unroll
      for (int c = 0; c < 3; ++c) {
        v2f a;
        a.x = sS[r * SS_STR + k0 * 3 + c];
        a.y = sS[r * SS_STR + (k0 + 1) * 3 + c];
        hacc[c][0] = __builtin_amdgcn_wmma_f32_16x16x4_f32(
            false, a, false, bb0, (short)0, hacc[c][0], false, false);
        hacc[c][1] = __builtin_amdgcn_wmma_f32_16x16x4_f32(
            false, a, false, bb1, (short)0, hacc[c][1], false, false);
      }
    }

    // ---- gate (sigmoid of channel norm), project Wsph2, reduce 32 ch ----
    float sx[8], sy[8], sz[8];
#pragma unroll
    for (int v = 0; v < 8; ++v) { sx[v] = 0.f; sy[v] = 0.f; sz[v] = 0.f; }
#pragma unroll
    for (int nt = 0; nt < 2; ++nt)
#pragma unroll
      for (int v = 0; v < 8; ++v) {
        float hx = hacc[0][nt][v] * 0.125f;   // / sqrt(64)
        float hy = hacc[1][nt][v] * 0.125f;
        float hz = hacc[2][nt][v] * 0.125f;
        float nrm = __builtin_amdgcn_sqrtf(hx * hx + hy * hy + hz * hz);
        float g = fast_sigmoid(nrm);
        float w = g * wsp2v[nt];
        sx[v] += hx * w; sy[v] += hy * w; sz[v] += hz * w;
      }
    butterfly16(sx); butterfly16(sy); butterfly16(sz);

    float av[8];
#pragma unroll
    for (int v = 0; v < 8; ++v) {
      // two 1/sqrt(32) factors -> 1/32 on |v|^2
      float s2 = (sx[v] * sx[v] + sy[v] * sy[v] + sz[v] * sz[v]) * 0.03125f;
      av[v] = s2 * so[v];
    }
    // rows 0-7 from lower half (lanes 0-7), rows 8-15 from upper (lanes 16-23)
    if (r < 8) {
      int m = (half << 3) | r;
      int row = a0 + m;
      if (row < N) atom_out[row] = pick8(av, r);
    }
  }
}

// one wave per molecule; batch_index is sorted, binary-search the span.
__global__ void __launch_bounds__(256)
seg_sum(const float* __restrict__ atom, const int* __restrict__ bidx,
        float* __restrict__ out, int N, int Bseg) {
  const int gw = (blockIdx.x * blockDim.x + threadIdx.x) >> 5;
  const int lane = threadIdx.x & 31;
  if (gw >= Bseg) return;
  int lo = 0, hi = N;
  while (lo < hi) { int mid = (lo + hi) >> 1; if (bidx[mid] < gw) lo = mid + 1; else hi = mid; }
  const int start = lo;
  hi = N;
  while (lo < hi) { int mid = (lo + hi) >> 1; if (bidx[mid] < gw + 1) lo = mid + 1; else hi = mid; }
  const int end = lo;
  float s = 0.f;
  for (int i = start + lane; i < end; i += 32) s += atom[i];
#pragma unroll
  for (int off = 1; off <= 16; off <<= 1) s += __shfl_xor(s, off, 32);
  if (lane == 0) out[gw] = s;
}

extern "C" void kernel_launch(void* const* d_in, const int* in_sizes, int n_in,
                              void* d_out, int out_size, void* d_ws, size_t ws_size,
                              hipStream_t stream) {
  const float* xs    = (const float*)d_in[0];
  const float* xsph  = (const float*)d_in[1];
  // d_in[2] = coord (unused by the reference)
  const int*   bidx  = (const int*)d_in[3];
  const float* W1    = (const float*)d_in[4];
  const float* b1    = (const float*)d_in[5];
  const float* W2    = (const float*)d_in[6];
  const float* b2    = (const float*)d_in[7];
  const float* Wsph1 = (const float*)d_in[8];
  const float* Wsph2 = (const float*)d_in[9];
  float* out = (float*)d_out;
  float* atom_ws = (float*)d_ws;  // N floats of scratch

  const int N = in_sizes[0] / 128;
  const int tiles = (N + 15) / 16;
  int blocks1 = (tiles + 7) / 8;
  if (blocks1 > 1024) blocks1 = 1024;
  if (blocks1 < 1) blocks1 = 1;
  const size_t shmem = (size_t)SMEM_FLOATS * sizeof(float);

  spatial_main<<<blocks1, 256, shmem, stream>>>(xs, xsph, W1, b1, W2, b2,
                                                Wsph1, Wsph2, atom_ws, N, tiles);

  const int Bseg = out_size;
  const int blocks2 = (Bseg + 7) / 8;
  seg_sum<<<blocks2, 256, 0, stream>>>(atom_ws, bidx, out, N, Bseg);
  (void)n_in; (void)ws_size;
}